// YoloLayer_43430709297348
// MI455X (gfx1250) — compile-verified
//
#include <hip/hip_runtime.h>
#include <hip/hip_bf16.h>
#include <stdint.h>
#include <stddef.h>

#ifndef __has_builtin
#define __has_builtin(x) 0
#endif

#define AS1 __attribute__((address_space(1)))
#define AS3 __attribute__((address_space(3)))

typedef unsigned int v4u __attribute__((ext_vector_type(4)));
typedef int          v8i __attribute__((ext_vector_type(8)));
typedef int          v4i __attribute__((ext_vector_type(4)));

#if __has_builtin(__builtin_amdgcn_global_load_async_to_lds_b32)
#define HAVE_ASYNC 1
#else
#define HAVE_ASYNC 0
#endif

#if __has_builtin(__builtin_amdgcn_tensor_load_to_lds)
#define HAVE_TDM 1
#else
#define HAVE_TDM 0
#endif

#if __has_builtin(__builtin_amdgcn_s_wait_asynccnt)
#define ASYNC_WAIT() __builtin_amdgcn_s_wait_asynccnt(0)
#elif HAVE_ASYNC
#define ASYNC_WAIT() asm volatile("s_wait_asynccnt 0" ::: "memory")
#else
#define ASYNC_WAIT() ((void)0)
#endif

// ---------------- problem geometry ----------------
#define BATCH   16
#define NCLS    80
#define N3      19200   // 80*80*3
#define N4      4800    // 40*40*3
#define N5      1200    // 20*20*3
#define N_ALL   25200
#define N_PAD   25344   // 99*256
#define CH_A    99
#define KTOP    200
#define FLAT    16000   // NCLS*KTOP
#define FLAT_PAD 16128  // 63*256
#define CH_B    63
#define IOU_T   0.6f

// Kernel A dynamic-LDS layout (byte offsets; no static LDS -> dynamic base == 0)
#define A_OFF_S    0        // 25344 f  (scores, padded)
#define A_OFF_RV   101376   // 256 f    (reduce values)
#define A_OFF_RI   102400   // 256 i    (reduce indices)
#define A_OFF_TV   103424   // 256 f    (top values)
#define A_OFF_TI   104448   // 256 i    (top indices)
#define A_OFF_BOX  105472   // 200*4 f  (raw params then decoded boxes)
#define A_OFF_KEEP 108672   // 256 i
#define SMEM_A     109696

// Kernel B dynamic-LDS layout
#define B_OFF_S    0        // 16128 f
#define B_OFF_RV   64512
#define B_OFF_RI   65536
#define B_OFF_TV   66560
#define B_OFF_TI   67584
#define B_OFF_CNT  68608
#define SMEM_B     68624

__constant__ float c_anch[9][2] = {
  {12.f,16.f},{19.f,36.f},{40.f,28.f},
  {36.f,75.f},{76.f,55.f},{72.f,146.f},
  {142.f,110.f},{192.f,243.f},{459.f,401.f}};

__device__ __forceinline__ float sigmoidf(float x) {
  return 1.0f / (1.0f + __expf(-x));
}

__device__ __forceinline__ float iouf(float ay0, float ax0, float ay1, float ax1,
                                      float by0, float bx0, float by1, float bx1) {
  float areaA = (ay1 - ay0) * (ax1 - ax0);
  float areaB = (by1 - by0) * (bx1 - bx0);
  float yy0 = fmaxf(ay0, by0), xx0 = fmaxf(ax0, bx0);
  float yy1 = fminf(ay1, by1), xx1 = fminf(ax1, bx1);
  float inter = fmaxf(yy1 - yy0, 0.f) * fmaxf(xx1 - xx0, 0.f);
  float uni = areaA + areaB - inter;
  return inter / fmaxf(uni, 1e-9f);
}

// Repeated argmax over LDS-resident scores. Each thread caches the argmax of
// its contiguous CH-element chunk; per round: one 256-wide LDS tree reduce +
// a single-chunk rescan by the owning thread. Ties -> smaller index (top_k order).
__device__ __forceinline__ void argmax_rounds(float* sS, float* rv, int* ri,
                                              float* tv, int* ti,
                                              int tid, int CH, int rounds) {
  const int lo = tid * CH, hi = lo + CH;
  float mv = -3e38f; int mi = lo;
  for (int i = lo; i < hi; ++i) { float v = sS[i]; if (v > mv) { mv = v; mi = i; } }
  for (int k = 0; k < rounds; ++k) {
    rv[tid] = mv; ri[tid] = mi;
    __syncthreads();
    for (int s = 128; s > 0; s >>= 1) {
      if (tid < s) {
        float a = rv[tid], b = rv[tid + s];
        int ia = ri[tid], ib = ri[tid + s];
        if (b > a || (b == a && ib < ia)) { rv[tid] = b; ri[tid] = ib; }
      }
      __syncthreads();
    }
    float bv = rv[0]; int bi = ri[0];
    if (tid == 0) { tv[k] = bv; ti[k] = bi; }
    if (bi >= lo && bi < hi) {           // owner invalidates + rescans its chunk
      sS[bi] = -3e38f;
      mv = -3e38f; mi = lo;
      for (int i = lo; i < hi; ++i) { float v = sS[i]; if (v > mv) { mv = v; mi = i; } }
    }
    __syncthreads();
  }
}

// ---------------------------------------------------------------------------
// Kernel A: one block per (class, batch). Recompute the class score column
// into LDS, top-200 via chunked argmax, async-gather the winners' raw box
// params into LDS, decode, greedy NMS, write candidate pool to workspace.
// ---------------------------------------------------------------------------
__global__ __launch_bounds__(256)
void yolo_topk_nms_kernel(const float* __restrict__ p3, const float* __restrict__ p4,
                          const float* __restrict__ p5,
                          float* __restrict__ nmsB, float* __restrict__ nmsS) {
  extern __shared__ char smem[];
  float* sS    = (float*)(smem + A_OFF_S);
  float* rv    = (float*)(smem + A_OFF_RV);
  int*   ri    = (int*)  (smem + A_OFF_RI);
  float* tv    = (float*)(smem + A_OFF_TV);
  int*   ti    = (int*)  (smem + A_OFF_TI);
  float* sBox  = (float*)(smem + A_OFF_BOX);
  int*   sKeep = (int*)  (smem + A_OFF_KEEP);

  const int tid = threadIdx.x;
  const int c = blockIdx.x;
  const int b = blockIdx.y;

  // ---- Phase 1: scores = sigmoid(obj)*sigmoid(cls_c)
  // Per-position layout [H,W,A,85] => element (n, ch) at flat 85*n + ch
  // (255 == 3*85), so NO div/mod in the hot loop. Masks at thresh 0 are identity.
  {
    const float* b3 = p3 + (size_t)b * N3 * 85;
    const float* b4 = p4 + (size_t)b * N4 * 85;
    const float* b5 = p5 + (size_t)b * N5 * 85;
    for (int n = tid; n < N3; n += 256) {
      size_t off = (size_t)n * 85;
      sS[n] = sigmoidf(b3[off + 4]) * sigmoidf(b3[off + 5 + c]);
    }
    for (int n = tid; n < N4; n += 256) {
      size_t off = (size_t)n * 85;
      sS[N3 + n] = sigmoidf(b4[off + 4]) * sigmoidf(b4[off + 5 + c]);
    }
    for (int n = tid; n < N5; n += 256) {
      size_t off = (size_t)n * 85;
      sS[N3 + N4 + n] = sigmoidf(b5[off + 4]) * sigmoidf(b5[off + 5 + c]);
    }
    for (int n = N_ALL + tid; n < N_PAD; n += 256) sS[n] = -3e38f;
  }
  __syncthreads();

  // ---- Phase 2: top-200
  argmax_rounds(sS, rv, ri, tv, ti, tid, CH_A, KTOP);

  // ---- Phase 3: gather winners' raw (tx,ty,tw,th) into LDS (async), decode.
  // Per-level branches give compile-time divisors (3, 80/40/20) -> mul-shift.
  int lvl = 0, a = 0, HW = 80, wq = 0, hq = 0;
  if (tid < KTOP) {
    int idx = ti[tid];
    int nn; const float* src;
    if (idx < N3) {
      lvl = 0; HW = 80; nn = idx;
      a = nn % 3; int hw = nn / 3; wq = hw % 80; hq = hw / 80;
      src = p3 + (size_t)b * N3 * 85;
    } else if (idx < N3 + N4) {
      lvl = 1; HW = 40; nn = idx - N3;
      a = nn % 3; int hw = nn / 3; wq = hw % 40; hq = hw / 40;
      src = p4 + (size_t)b * N4 * 85;
    } else {
      lvl = 2; HW = 20; nn = idx - N3 - N4;
      a = nn % 3; int hw = nn / 3; wq = hw % 20; hq = hw / 20;
      src = p5 + (size_t)b * N5 * 85;
    }
    const float* g = src + (size_t)nn * 85;
#if HAVE_ASYNC
    // 4 consecutive floats, 4B-aligned (85*4 stride) -> 4x async b32 to LDS.
    // INST_OFFSET is added to both global and LDS addresses (ISA 08 §4.4).
    AS1 int* gp = (AS1 int*)(uintptr_t)g;
    AS3 int* lp = (AS3 int*)(size_t)(unsigned)(A_OFF_BOX + tid * 16);
    __builtin_amdgcn_global_load_async_to_lds_b32(gp, lp, 0, 0);
    __builtin_amdgcn_global_load_async_to_lds_b32(gp, lp, 4, 0);
    __builtin_amdgcn_global_load_async_to_lds_b32(gp, lp, 8, 0);
    __builtin_amdgcn_global_load_async_to_lds_b32(gp, lp, 12, 0);
#else
    sBox[tid * 4 + 0] = g[0]; sBox[tid * 4 + 1] = g[1];
    sBox[tid * 4 + 2] = g[2]; sBox[tid * 4 + 3] = g[3];
#endif
  }
  ASYNC_WAIT();
  __syncthreads();

  if (tid < KTOP) {
    float t0 = sBox[tid * 4 + 0], t1 = sBox[tid * 4 + 1];
    float t2 = sBox[tid * 4 + 2], t3 = sBox[tid * 4 + 3];
    float inv = 1.0f / (float)HW;
    float cx = ((float)wq + sigmoidf(t0)) * inv;
    float cy = ((float)hq + sigmoidf(t1)) * inv;
    int ai = lvl * 3 + a;
    // stride*HW == 640 for all three levels
    float bw = __expf(t2) * c_anch[ai][0] * (1.0f / 640.0f);
    float bh = __expf(t3) * c_anch[ai][1] * (1.0f / 640.0f);
    sBox[tid * 4 + 0] = cy - 0.5f * bh;
    sBox[tid * 4 + 1] = cx - 0.5f * bw;
    sBox[tid * 4 + 2] = cy + 0.5f * bh;
    sBox[tid * 4 + 3] = cx + 0.5f * bw;
    sKeep[tid] = 1;
  }
  __syncthreads();

  // ---- Phase 4: greedy NMS (thread j owns candidate j; i scans in rank order)
  float my0 = 0, my1 = 0, my2 = 0, my3 = 0; int myKeep = 0;
  if (tid < KTOP) {
    my0 = sBox[tid * 4 + 0]; my1 = sBox[tid * 4 + 1];
    my2 = sBox[tid * 4 + 2]; my3 = sBox[tid * 4 + 3];
    myKeep = 1;
  }
  for (int i = 0; i < KTOP; ++i) {
    __syncthreads();
    if (sKeep[i] && tid > i && tid < KTOP && myKeep) {
      float u = iouf(sBox[i * 4 + 0], sBox[i * 4 + 1], sBox[i * 4 + 2], sBox[i * 4 + 3],
                     my0, my1, my2, my3);
      if (u > IOU_T) { myKeep = 0; sKeep[tid] = 0; }
    }
  }

  // ---- Phase 5: emit candidate pool (boxes always; scores zeroed if suppressed)
  if (tid < KTOP) {
    size_t base = ((size_t)(b * NCLS + c)) * KTOP + tid;
    nmsS[base] = myKeep ? tv[tid] : 0.0f;
    float* ob = nmsB + base * 4;
    ob[0] = my0; ob[1] = my1; ob[2] = my2; ob[3] = my3;
  }
}

// ---------------------------------------------------------------------------
// Kernel B: one block per batch. TDM-DMA the 16000-float candidate pool into
// LDS, top-200, gather boxes, emit final outputs.
// ---------------------------------------------------------------------------
__global__ __launch_bounds__(256)
void yolo_final_topk_kernel(const float* __restrict__ nmsS, const float* __restrict__ nmsB,
                            float* __restrict__ out) {
  extern __shared__ char smem[];
  float* sS  = (float*)(smem + B_OFF_S);
  float* rv  = (float*)(smem + B_OFF_RV);
  int*   ri  = (int*)  (smem + B_OFF_RI);
  float* tv  = (float*)(smem + B_OFF_TV);
  int*   ti  = (int*)  (smem + B_OFF_TI);
  int*   cnt = (int*)  (smem + B_OFF_CNT);

  const int tid = threadIdx.x;
  const int b = blockIdx.x;
  const float* src = nmsS + (size_t)b * FLAT;

#if HAVE_TDM
  if ((tid >> 5) == 0) {   // wave 0 issues the DMA (uniform branch)
    unsigned long long ga = (unsigned long long)(uintptr_t)src;
    // D# group 0: count=1 | lds_addr | global_addr[56:0] | type=2 (ISA 08 §8.3)
    v4u g0 = { 1u,
               (unsigned)B_OFF_S,
               (unsigned)(ga & 0xffffffffull),
               (unsigned)((ga >> 32) & 0x1ffffffull) | (2u << 30) };
    // D# group 1: data_size=4B; 2-D tensor dim0=16000, dim1=1; tile 16000x1;
    // dim0_stride=16000 (ISA 08 §8.4)
    v8i g1 = { (int)(2u << 16),
               (int)(((unsigned)FLAT & 0xffffu) << 16),
               (int)(((unsigned)FLAT >> 16) | (1u << 16)),
               (int)(((unsigned)FLAT & 0xffffu) << 16),
               1,
               (int)FLAT,
               0, 0 };
    v4i z4 = { 0, 0, 0, 0 };
#if __clang_major__ >= 23
    v8i z8 = { 0, 0, 0, 0, 0, 0, 0, 0 };
    __builtin_amdgcn_tensor_load_to_lds(g0, g1, z4, z4, z8, 0);
#else
    __builtin_amdgcn_tensor_load_to_lds(g0, g1, z4, z4, 0);
#endif
    __builtin_amdgcn_s_wait_tensorcnt(0);
  }
#else
  for (int i = tid; i < FLAT; i += 256) sS[i] = src[i];
#endif
  for (int i = FLAT + tid; i < FLAT_PAD; i += 256) sS[i] = -3e38f;  // padding
  __syncthreads();

  argmax_rounds(sS, rv, ri, tv, ti, tid, CH_B, KTOP);

  float* oB = out;                         // [16,200,4]
  float* oS = out + BATCH * KTOP * 4;      // [16,200]
  float* oC = oS + BATCH * KTOP;           // [16,200]
  float* oN = oC + BATCH * KTOP;           // [16]
  if (tid == 0) *cnt = 0;
  __syncthreads();
  if (tid < KTOP) {
    int fi = ti[tid]; float v = tv[tid];
    int cls = fi / KTOP;                   // constant divisor -> mul-shift
    const float* bb = nmsB + (((size_t)b * FLAT) + fi) * 4;
    float* ob = oB + ((size_t)(b * KTOP + tid)) * 4;
    ob[0] = bb[0]; ob[1] = bb[1]; ob[2] = bb[2]; ob[3] = bb[3];
    oS[b * KTOP + tid] = v;
    oC[b * KTOP + tid] = (float)cls;
    if (v > 0.0f) atomicAdd(cnt, 1);
  }
  __syncthreads();
  if (tid == 0) oN[b] = (float)(*cnt);
}

// ---------------------------------------------------------------------------
extern "C" void kernel_launch(void* const* d_in, const int* in_sizes, int n_in,
                              void* d_out, int out_size, void* d_ws, size_t ws_size,
                              hipStream_t stream) {
  const float* p3 = (const float*)d_in[0];
  const float* p4 = (const float*)d_in[1];
  const float* p5 = (const float*)d_in[2];
  float* out = (float*)d_out;

  // Workspace: nmsB [B,C,K,4] (4.096 MB) then nmsS [B,C,K] (1.024 MB)
  float* nmsB = (float*)d_ws;
  float* nmsS = nmsB + (size_t)BATCH * NCLS * KTOP * 4;

  // >64KB dynamic LDS opt-in (no-op if already set; not a stream operation)
  (void)hipFuncSetAttribute(reinterpret_cast<const void*>(yolo_topk_nms_kernel),
                            hipFuncAttributeMaxDynamicSharedMemorySize, SMEM_A);
  (void)hipFuncSetAttribute(reinterpret_cast<const void*>(yolo_final_topk_kernel),
                            hipFuncAttributeMaxDynamicSharedMemorySize, SMEM_B);

  yolo_topk_nms_kernel<<<dim3(NCLS, BATCH), 256, SMEM_A, stream>>>(p3, p4, p5, nmsB, nmsS);
  yolo_final_topk_kernel<<<dim3(BATCH), 256, SMEM_B, stream>>>(nmsS, nmsB, out);
}